// MultiHeadedAttention_2_18631568130097
// MI455X (gfx1250) — compile-verified
//
#include <hip/hip_runtime.h>

typedef __attribute__((ext_vector_type(2))) float v2f;
typedef __attribute__((ext_vector_type(4))) float v4f;
typedef __attribute__((ext_vector_type(8))) float v8f;

#define BB 8
#define NUM 128
#define HH 8
#define DK 64
#define LL (NUM * NUM)          // 16384
#define LG (LL / 16)            // 1024 tile groups per (b)
#define EPSV 1e-8f

// select element idx (0..7) out of a v8f accumulator (cndmask chain, no scratch)
__device__ __forceinline__ float pick8(v8f v, unsigned i) {
    float r = v[0];
    r = (i == 1u) ? v[1] : r;
    r = (i == 2u) ? v[2] : r;
    r = (i == 3u) ? v[3] : r;
    r = (i == 4u) ? v[4] : r;
    r = (i == 5u) ? v[5] : r;
    r = (i == 6u) ? v[6] : r;
    r = (i == 7u) ? v[7] : r;
    return r;
}

// ---------------------------------------------------------------------------
// Kernel A: per (b, h, 16-row tile) compute diag(QK^T), diag(QQ^T), diag(KK^T)
// via chained v_wmma_f32_16x16x4_f32, then p = relu(dot / max(|q||k|, eps)).
// attn written in (B, i, j, H) layout.
// grid = B*LG blocks of 256 threads (8 waves = 8 heads).
// ---------------------------------------------------------------------------
__global__ void __launch_bounds__(256)
cos_attn_wmma(const float* __restrict__ q, const float* __restrict__ k,
              float* __restrict__ attn) {
    const unsigned h    = threadIdx.x >> 5;      // wave id = head
    const unsigned lane = threadIdx.x & 31u;
    const unsigned lg   = blockIdx.x & (LG - 1u);
    const unsigned b    = blockIdx.x >> 10;

    const unsigned r  = lane & 15u;              // tile row (M and N)
    const unsigned c2 = (lane >> 4) << 1;        // K sub-offset: 0 or 2

    // element offset of (b, l = lg*16 + r, h, d = c2) in (B, L, H, DK) layout
    const size_t base = (((size_t)b * LL + (size_t)lg * 16u + r) * (HH * DK))
                      + (size_t)h * DK + c2;
    const float* qp = q + base;
    const float* kp = k + base;

    v8f dQK = {}; v8f dQQ = {}; v8f dKK = {};
#pragma unroll
    for (int s = 0; s < 16; ++s) {               // K = 4 per step, d = 4s..4s+3
        v2f av = *(const v2f*)(qp + 4 * s);      // serves as A (16x4) and B (4x16)
        v2f bv = *(const v2f*)(kp + 4 * s);
        dQK = __builtin_amdgcn_wmma_f32_16x16x4_f32(false, av, false, bv,
                                                    (short)0, dQK, false, false);
        dQQ = __builtin_amdgcn_wmma_f32_16x16x4_f32(false, av, false, av,
                                                    (short)0, dQQ, false, false);
        dKK = __builtin_amdgcn_wmma_f32_16x16x4_f32(false, bv, false, bv,
                                                    (short)0, dKK, false, false);
    }

    // Diagonal (M,M): lanes 0-7 hold M=lane in VGPR=lane; lanes 24-31 hold
    // M=lane-16 in VGPR=lane-24. Lanes 8-23 hold no diagonal element.
    const bool     holds = (lane < 8u) || (lane >= 24u);
    const unsigned idx   = (lane < 8u) ? lane : (lane - 24u);
    const unsigned m     = (lane < 8u) ? lane : (lane - 16u);
    if (holds) {
        const float dot = pick8(dQK, idx);
        const float qq  = pick8(dQQ, idx);
        const float kk  = pick8(dKK, idx);
        float denom = sqrtf(qq * kk);            // = |q|*|k|
        denom = fmaxf(denom, EPSV);
        const float p = fmaxf(dot / denom, 0.0f);
        const unsigned l = lg * 16u + m;
        const unsigned i = l >> 7;               // l / NUM
        const unsigned j = l & (NUM - 1u);
        attn[((((size_t)b * NUM + i) * NUM + j) * HH) + h] = p;
    }
}

// ---------------------------------------------------------------------------
// Kernel 0: clear the global 128-entry column mask
// ---------------------------------------------------------------------------
__global__ void clear_mask(float* __restrict__ cmask) {
    cmask[threadIdx.x] = 0.0f;
}

// ---------------------------------------------------------------------------
// Kernel B: one thread per (b, i, h) row; stable top-4 over j of
// attn[b, i, j, h]; mark colmask[j] = 1 for the winners.
// ---------------------------------------------------------------------------
__global__ void topk_mask(const float* __restrict__ attn,
                          float* __restrict__ cmask) {
    const int t = blockIdx.x * blockDim.x + threadIdx.x;
    if (t >= BB * NUM * HH) return;
    const int h = t & (HH - 1);
    const int i = (t >> 3) & (NUM - 1);
    const int b = t >> 10;
    const float* row = attn + (((size_t)b * NUM + i) * NUM) * HH + h;

    float v0 = -3.4e38f, v1 = v0, v2 = v0, v3 = v0;
    int   i0 = 0, i1 = 0, i2 = 0, i3 = 0;
#pragma unroll 4
    for (int j = 0; j < NUM; ++j) {
        const float x = row[(size_t)j * HH];
        if (x > v3) {                            // strict '>' => earlier index wins ties
            v3 = x; i3 = j;
            if (v3 > v2) { float tv = v2; v2 = v3; v3 = tv; int ti = i2; i2 = i3; i3 = ti; }
            if (v2 > v1) { float tv = v1; v1 = v2; v2 = tv; int ti = i1; i1 = i2; i2 = ti; }
            if (v1 > v0) { float tv = v0; v0 = v1; v1 = tv; int ti = i0; i0 = i1; i1 = ti; }
        }
    }
    cmask[i0] = 1.0f;
    cmask[i1] = 1.0f;
    cmask[i2] = 1.0f;
    cmask[i3] = 1.0f;
}

// ---------------------------------------------------------------------------
// Kernel C: out[b,i,j,:] = attn[b,i,j,:] * roi[b,i,j] * cmask[j]
// one thread per (b,i,j), 32B vector load/store over the 8 heads.
// ---------------------------------------------------------------------------
__global__ void __launch_bounds__(256)
finalize(const float* __restrict__ attn, const float* __restrict__ roi,
         const float* __restrict__ cmask, float* __restrict__ out) {
    const int t = blockIdx.x * blockDim.x + threadIdx.x;
    if (t >= BB * NUM * NUM) return;
    const int j = t & (NUM - 1);
    const float s = roi[t] * cmask[j];
    const v4f* ap = (const v4f*)(attn + (size_t)t * HH);
    v4f a0 = ap[0], a1 = ap[1];
    a0.x *= s; a0.y *= s; a0.z *= s; a0.w *= s;
    a1.x *= s; a1.y *= s; a1.z *= s; a1.w *= s;
    v4f* op = (v4f*)(out + (size_t)t * HH);
    op[0] = a0;
    op[1] = a1;
}

extern "C" void kernel_launch(void* const* d_in, const int* in_sizes, int n_in,
                              void* d_out, int out_size, void* d_ws, size_t ws_size,
                              hipStream_t stream) {
    const float* query = (const float*)d_in[0];  // (8,128,128,512) f32
    const float* key   = (const float*)d_in[1];  // (8,128,128,512) f32
    const float* roi   = (const float*)d_in[2];  // (8,128,128)     f32
    float* out = (float*)d_out;                  // (8,128,128,8)   f32

    // workspace layout: [0, 1KB): column mask (128 floats, padded)
    //                   [1KB, 1KB + 4MB): attn buffer (B,i,j,H) f32
    float* cmask = (float*)d_ws;
    float* attn  = (float*)d_ws + 256;

    clear_mask<<<1, 128, 0, stream>>>(cmask);

    cos_attn_wmma<<<BB * LG, 256, 0, stream>>>(query, key, attn);   // 8192 blocks

    topk_mask<<<(BB * NUM * HH + 255) / 256, 256, 0, stream>>>(attn, cmask);

    finalize<<<(BB * NUM * NUM + 255) / 256, 256, 0, stream>>>(attn, roi, cmask, out);
}